// GroupedQueryAttention_7241314861176
// MI455X (gfx1250) — compile-verified
//
#include <hip/hip_runtime.h>
#include <hip/hip_bf16.h>
#include <math.h>

// ---------------------------------------------------------------------------
// GQA attention w/ 2D-RoPE for MI455X (gfx1250, wave32, WMMA).
// Matrix math: V_WMMA_F32_16X16X4_F32 (exact f32 matrix op).
// Staging: GLOBAL_LOAD_ASYNC_TO_LDS_B128 (ASYNCcnt) when available,
//          register round-trip fallback otherwise.
// ---------------------------------------------------------------------------

typedef float v2f __attribute__((ext_vector_type(2)));
typedef float v8f __attribute__((ext_vector_type(8)));
typedef int   v4i __attribute__((vector_size(16)));

#define B_SZ    32
#define T_SEQ   576
#define D_MODEL 1024
#define NQH     16
#define NKVH    4
#define HD      64
#define QKV_N   1536             // H*HD + 2*G*HD
#define BT      (B_SZ * T_SEQ)   // 18432
#define SCALE_ATTN 0.125f        // 1/sqrt(64)

#define KBLK   64                // K-block staged in LDS per GEMM iteration
#define LDS_P  68                // padded LDS row stride (A-frag reads conflict-free)
#define LDS_PV 72                // padded LDS row stride for V-phase reads

// ---- CDNA5 async global->LDS path (probe-guarded) --------------------------
#if defined(__has_builtin)
#if __has_builtin(__builtin_amdgcn_global_load_async_to_lds_b128)
#define HAS_ASYNC 1
#endif
#endif
#ifndef HAS_ASYNC
#define HAS_ASYNC 0
#endif

#if HAS_ASYNC
static __device__ __forceinline__ void async_cp16(void* lds, const void* g) {
    __builtin_amdgcn_global_load_async_to_lds_b128(
        (__attribute__((address_space(1))) v4i*)(g),
        (__attribute__((address_space(3))) v4i*)(lds), 0, 0);
}
#if __has_builtin(__builtin_amdgcn_s_wait_asynccnt)
#define ASYNC_WAIT0() __builtin_amdgcn_s_wait_asynccnt(0)
#else
#define ASYNC_WAIT0() asm volatile("s_wait_asynccnt 0x0" ::: "memory")
#endif
#endif

static __device__ __forceinline__ v8f wmma_f32(v2f a, v2f b, v8f c) {
    // 8 args: (neg_a, A, neg_b, B, c_mod, C, reuse_a, reuse_b)
    return __builtin_amdgcn_wmma_f32_16x16x4_f32(false, a, false, b, (short)0, c,
                                                 false, false);
}

// ---------------------------------------------------------------------------
// C[M x N] = A[M x K] @ W[N x K]^T   (row-major, f32, WMMA 16x16x4)
// block = 256 threads = 8 waves; block tile = 128(M) x 64(N); wave = 16 x 64.
// Double-buffered LDS staging of A (128xKBLK) and B (64xKBLK) tiles.
// ---------------------------------------------------------------------------
__global__ __launch_bounds__(256) void gemm_nt_wmma(
    const float* __restrict__ A, const float* __restrict__ W,
    float* __restrict__ C, int K, int lda, int ldb, int ldc)
{
    __shared__ float sA[2][128 * LDS_P];   // 2 * 34816 B
    __shared__ float sB[2][64 * LDS_P];    // 2 * 17408 B

    const int tid  = threadIdx.x;
    const int lane = tid & 31;
    const int wave = tid >> 5;           // 0..7
    const int half = lane >> 4;          // K-pair select
    const int l16  = lane & 15;

    const int n0 = blockIdx.x * 64;
    const int m0 = blockIdx.y * 128;
    const int mw = wave * 16;

    v8f acc0 = {}, acc1 = {}, acc2 = {}, acc3 = {};

    auto computeBlk = [&](int cur) {
        const float* pA = &sA[cur][mw * LDS_P];
        const float* pB = &sB[cur][0];
#pragma unroll
        for (int k = 0; k < KBLK; k += 4) {
            const int kk = k + 2 * half;
            v2f a  = *(const v2f*)(pA + (size_t)l16 * LDS_P + kk);
            v2f b0 = *(const v2f*)(pB + (size_t)(l16     ) * LDS_P + kk);
            v2f b1 = *(const v2f*)(pB + (size_t)(l16 + 16) * LDS_P + kk);
            v2f b2 = *(const v2f*)(pB + (size_t)(l16 + 32) * LDS_P + kk);
            v2f b3 = *(const v2f*)(pB + (size_t)(l16 + 48) * LDS_P + kk);
            acc0 = wmma_f32(a, b0, acc0);
            acc1 = wmma_f32(a, b1, acc1);
            acc2 = wmma_f32(a, b2, acc2);
            acc3 = wmma_f32(a, b3, acc3);
        }
    };

    const int NB = K / KBLK;

#if HAS_ASYNC
    // ---- async global->LDS staging (no VGPR round-trip) --------------------
    auto issueAB = [&](int buf, int k0) {
#pragma unroll
        for (int i = 0; i < 8; ++i) {           // A: 128 rows x 16 float4
            const int id = tid + i * 256;
            const int r = id >> 4, c = (id & 15) << 2;
            async_cp16(&sA[buf][r * LDS_P + c],
                       A + (size_t)(m0 + r) * lda + k0 + c);
        }
#pragma unroll
        for (int i = 0; i < 4; ++i) {           // B: 64 rows x 16 float4
            const int id = tid + i * 256;
            const int r = id >> 4, c = (id & 15) << 2;
            async_cp16(&sB[buf][r * LDS_P + c],
                       W + (size_t)(n0 + r) * ldb + k0 + c);
        }
    };

    issueAB(0, 0);
    ASYNC_WAIT0();
    __syncthreads();
    for (int kb = 0; kb < NB; ++kb) {
        const int cur = kb & 1;
        if (kb + 1 < NB) issueAB(cur ^ 1, (kb + 1) * KBLK);
        computeBlk(cur);
        if (kb + 1 < NB) ASYNC_WAIT0();
        __syncthreads();
    }
#else
    // ---- fallback: loads into VGPRs, then ds stores ------------------------
    float4 ra[8]; float4 rb[4];
    auto loadAB = [&](int k0) {
#pragma unroll
        for (int i = 0; i < 8; ++i) {
            const int id = tid + i * 256;
            const int r = id >> 4, c = (id & 15) << 2;
            ra[i] = *(const float4*)(A + (size_t)(m0 + r) * lda + k0 + c);
        }
#pragma unroll
        for (int i = 0; i < 4; ++i) {
            const int id = tid + i * 256;
            const int r = id >> 4, c = (id & 15) << 2;
            rb[i] = *(const float4*)(W + (size_t)(n0 + r) * ldb + k0 + c);
        }
    };
    auto stageAB = [&](int buf) {
#pragma unroll
        for (int i = 0; i < 8; ++i) {
            const int id = tid + i * 256;
            const int r = id >> 4, c = (id & 15) << 2;
            *(float4*)(&sA[buf][r * LDS_P + c]) = ra[i];
        }
#pragma unroll
        for (int i = 0; i < 4; ++i) {
            const int id = tid + i * 256;
            const int r = id >> 4, c = (id & 15) << 2;
            *(float4*)(&sB[buf][r * LDS_P + c]) = rb[i];
        }
    };

    loadAB(0);
    stageAB(0);
    __syncthreads();
    for (int kb = 0; kb < NB; ++kb) {
        const int cur = kb & 1;
        if (kb + 1 < NB) loadAB((kb + 1) * KBLK);
        computeBlk(cur);
        __syncthreads();
        if (kb + 1 < NB) stageAB(cur ^ 1);
        __syncthreads();
    }
#endif

    // C/D layout: VGPR v -> row (v + 8*half), lane&15 -> column
#pragma unroll
    for (int v = 0; v < 8; ++v) {
        const int row = m0 + mw + v + 8 * half;
        float* crow = C + (size_t)row * ldc + n0 + l16;
        crow[0]  = acc0[v];
        crow[16] = acc1[v];
        crow[32] = acc2[v];
        crow[48] = acc3[v];
    }
}

// ---------------------------------------------------------------------------
// In-place 2D RoPE on q (cols 0..1023) and k (cols 1024..1279) of qkv buffer.
// ---------------------------------------------------------------------------
__global__ __launch_bounds__(256) void rope2d_kernel(float* __restrict__ qkv)
{
    const int total = BT * (NQH + NKVH) * 16;
    int idx = blockIdx.x * blockDim.x + threadIdx.x;
    if (idx >= total) return;

    const int j    = idx & 15;
    const int head = (idx >> 4) % (NQH + NKVH);
    const int bt   = idx / (16 * (NQH + NKVH));
    const int t    = bt % T_SEQ;

    const int colbase = (head < NQH) ? head * HD : D_MODEL + (head - NQH) * HD;
    float* p = qkv + (size_t)bt * QKV_N + colbase;

    const float inv_freq = __powf(10000.0f, -(float)j / 16.0f);
    const float tx = (float)(t / 24) * inv_freq;   // gx
    const float ty = (float)(t % 24) * inv_freq;   // gy
    float sx, cx, sy, cy;
    __sincosf(tx, &sx, &cx);
    __sincosf(ty, &sy, &cy);

    const float x1 = p[j], x2 = p[16 + j], y1 = p[32 + j], y2 = p[48 + j];
    p[j]      = x1 * cx - x2 * sx;
    p[16 + j] = x1 * sx + x2 * cx;
    p[32 + j] = y1 * cy - y2 * sy;
    p[48 + j] = y1 * sy + y2 * cy;
}

// ---------------------------------------------------------------------------
// Attention: one block = (b, h, 64 q-rows). 4 waves x 16 q-rows each.
// 64 x 576 score panel in LDS (144 KB) + double-buffered 64-row K/V chunk
// (36 KB) -> 180 KB of the 320 KB WGP LDS.
// ---------------------------------------------------------------------------
__global__ __launch_bounds__(128) void attn_wmma_kernel(
    const float* __restrict__ qkv, float* __restrict__ attn_out)
{
    __shared__ float sP[64 * T_SEQ];          // 147456 B
    __shared__ float sKV[2][64 * LDS_PV];     //  36864 B

    const int tid  = threadIdx.x;
    const int lane = tid & 31;
    const int wave = tid >> 5;                // 0..3
    const int half = lane >> 4;
    const int l16  = lane & 15;

    const int bid = blockIdx.x;
    const int qb  = bid % (T_SEQ / 64);
    const int h   = (bid / (T_SEQ / 64)) % NQH;
    const int b   = bid / ((T_SEQ / 64) * NQH);
    const int g   = h >> 2;

    const float* Qbase = qkv + (size_t)b * T_SEQ * QKV_N + h * HD;
    const float* Kbase = qkv + (size_t)b * T_SEQ * QKV_N + D_MODEL + g * HD;
    const float* Vbase = qkv + (size_t)b * T_SEQ * QKV_N + D_MODEL + NKVH * HD + g * HD;

    const int qloc = wave * 16;
    const int q0   = qb * 64 + qloc;

#if HAS_ASYNC
    auto issueKV = [&](int buf, const float* base, int row0, int stride) {
#pragma unroll
        for (int i = 0; i < 8; ++i) {         // 64 rows x 16 float4 / 128 thr
            const int id = tid + i * 128;
            const int r = id >> 4, c = (id & 15) << 2;
            async_cp16(&sKV[buf][r * stride + c],
                       base + (size_t)(row0 + r) * QKV_N + c);
        }
    };
#else
    float4 rkv[8];
    auto loadKV = [&](const float* base, int row0) {
#pragma unroll
        for (int i = 0; i < 8; ++i) {
            const int id = tid + i * 128;
            const int r = id >> 4, c = (id & 15) << 2;
            rkv[i] = *(const float4*)(base + (size_t)(row0 + r) * QKV_N + c);
        }
    };
    auto stageKV = [&](int buf, int stride) {
#pragma unroll
        for (int i = 0; i < 8; ++i) {
            const int id = tid + i * 128;
            const int r = id >> 4, c = (id & 15) << 2;
            *(float4*)(&sKV[buf][r * stride + c]) = rkv[i];
        }
    };
#endif

    // ---- preload Q A-fragments (loop-invariant across all key tiles) -------
    const float* Qrow = Qbase + (size_t)(q0 + l16) * QKV_N;
    v2f qa[16];
#pragma unroll
    for (int d = 0; d < 16; ++d)
        qa[d] = *(const v2f*)(Qrow + d * 4 + 2 * half);

    const int NC = T_SEQ / 64;   // 9 chunks

    // ---- scores: S[64 x 576] panel, K staged in 64-row chunks --------------
    auto scoreChunk = [&](int cur, int c) {
#pragma unroll
        for (int kt = 0; kt < 64; kt += 16) {
            const float* Krow = &sKV[cur][(kt + l16) * LDS_P + 2 * half];
            v8f acc = {};
#pragma unroll
            for (int d = 0; d < 16; ++d) {
                v2f bb = *(const v2f*)(Krow + d * 4);
                acc = wmma_f32(qa[d], bb, acc);
            }
#pragma unroll
            for (int v = 0; v < 8; ++v)
                sP[(qloc + v + 8 * half) * T_SEQ + c * 64 + kt + l16] = acc[v];
        }
    };

#if HAS_ASYNC
    issueKV(0, Kbase, 0, LDS_P);
    ASYNC_WAIT0();
    __syncthreads();
    for (int c = 0; c < NC; ++c) {
        const int cur = c & 1;
        if (c + 1 < NC) issueKV(cur ^ 1, Kbase, (c + 1) * 64, LDS_P);
        scoreChunk(cur, c);
        if (c + 1 < NC) ASYNC_WAIT0();
        __syncthreads();
    }
#else
    loadKV(Kbase, 0);
    stageKV(0, LDS_P);
    __syncthreads();
    for (int c = 0; c < NC; ++c) {
        const int cur = c & 1;
        if (c + 1 < NC) loadKV(Kbase, (c + 1) * 64);
        scoreChunk(cur, c);
        __syncthreads();
        if (c + 1 < NC) stageKV(cur ^ 1, LDS_P);
        __syncthreads();
    }
#endif

    // ---- softmax: one thread per row (scale folded into exp) ---------------
    if (tid < 64) {
        float* row = sP + tid * T_SEQ;
        float mx = -3.4e38f;
        for (int i = 0; i < T_SEQ; ++i) mx = fmaxf(mx, row[i]);
        float sum = 0.0f;
        for (int i = 0; i < T_SEQ; ++i) {
            float e = __expf((row[i] - mx) * SCALE_ATTN);
            row[i] = e;
            sum += e;
        }
        const float inv = 1.0f / sum;
        for (int i = 0; i < T_SEQ; ++i) row[i] *= inv;
    }
    __syncthreads();

    // ---- O[16 x 64] = P[16 x 576] @ V[576 x 64], V staged in chunks --------
    v8f o0 = {}, o1 = {}, o2 = {}, o3 = {};
    auto pvChunk = [&](int cur, int c) {
        const float* Prow = sP + (size_t)(qloc + l16) * T_SEQ + c * 64;
#pragma unroll
        for (int k = 0; k < 64; k += 4) {
            const int kk = k + 2 * half;
            v2f a = *(const v2f*)(Prow + kk);
            const float* v0 = &sKV[cur][kk * LDS_PV];
            const float* v1 = v0 + LDS_PV;
            v2f b0; b0.x = v0[l16];      b0.y = v1[l16];
            v2f b1; b1.x = v0[l16 + 16]; b1.y = v1[l16 + 16];
            v2f b2; b2.x = v0[l16 + 32]; b2.y = v1[l16 + 32];
            v2f b3; b3.x = v0[l16 + 48]; b3.y = v1[l16 + 48];
            o0 = wmma_f32(a, b0, o0);
            o1 = wmma_f32(a, b1, o1);
            o2 = wmma_f32(a, b2, o2);
            o3 = wmma_f32(a, b3, o3);
        }
    };

#if HAS_ASYNC
    issueKV(0, Vbase, 0, LDS_PV);
    ASYNC_WAIT0();
    __syncthreads();
    for (int c = 0; c < NC; ++c) {
        const int cur = c & 1;
        if (c + 1 < NC) issueKV(cur ^ 1, Vbase, (c + 1) * 64, LDS_PV);
        pvChunk(cur, c);
        if (c + 1 < NC) ASYNC_WAIT0();
        __syncthreads();
    }
#else
    loadKV(Vbase, 0);
    stageKV(0, LDS_PV);
    __syncthreads();
    for (int c = 0; c < NC; ++c) {
        const int cur = c & 1;
        if (c + 1 < NC) loadKV(Vbase, (c + 1) * 64);
        pvChunk(cur, c);
        __syncthreads();
        if (c + 1 < NC) stageKV(cur ^ 1, LDS_PV);
        __syncthreads();
    }
#endif

    float* Obase = attn_out + ((size_t)b * T_SEQ + q0) * D_MODEL + h * HD;
#pragma unroll
    for (int v = 0; v < 8; ++v) {
        float* orow = Obase + (size_t)(v + 8 * half) * D_MODEL + l16;
        orow[0]  = o0[v];
        orow[16] = o1[v];
        orow[32] = o2[v];
        orow[48] = o3[v];
    }
}

// ---------------------------------------------------------------------------
extern "C" void kernel_launch(void* const* d_in, const int* in_sizes, int n_in,
                              void* d_out, int out_size, void* d_ws, size_t ws_size,
                              hipStream_t stream)
{
    const float* x     = (const float*)d_in[0];   // (B, T, D)
    const float* w_qkv = (const float*)d_in[1];   // (1536, 1024)
    const float* w_o   = (const float*)d_in[2];   // (1024, 1024)
    float* out = (float*)d_out;                   // (B, T, 1024)

    float* qkv  = (float*)d_ws;                   // BT x 1536
    float* attn = qkv + (size_t)BT * QKV_N;       // BT x 1024

    // 1) qkv = x @ w_qkv^T
    gemm_nt_wmma<<<dim3(QKV_N / 64, BT / 128), 256, 0, stream>>>(
        x, w_qkv, qkv, D_MODEL, D_MODEL, D_MODEL, QKV_N);

    // 2) 2D RoPE in-place on q and k sections
    {
        const int total = BT * (NQH + NKVH) * 16;
        rope2d_kernel<<<(total + 255) / 256, 256, 0, stream>>>(qkv);
    }

    // 3) attention
    attn_wmma_kernel<<<B_SZ * NQH * (T_SEQ / 64), 128, 0, stream>>>(qkv, attn);

    // 4) out = attn @ w_o^T
    gemm_nt_wmma<<<dim3(D_MODEL / 64, BT / 128), 256, 0, stream>>>(
        attn, w_o, out, D_MODEL, D_MODEL, D_MODEL, D_MODEL);
}